// TimeConv_46926812676739
// MI455X (gfx1250) — compile-verified
//
#include <hip/hip_runtime.h>
#include <hip/hip_bf16.h>
#include <math.h>

// ---- problem constants (from reference) ----
#define B_DIM   8192
#define T_DIM   64
#define E_DIM   64
#define C_SZ    20          // CONV_SIZE
#define W_SZ    30          // WINDOW_SIZE
#define BEGIN   15          // T - (C+W-1)
#define LAST_T0 34          // "last" row for window w is x[:, 34+w, :]
#define OUT_F   320         // 5*64 features per (b,w)
#define NCOL    (W_SZ * E_DIM)   // 1920 rank columns

typedef __attribute__((ext_vector_type(2))) float v2f;
typedef __attribute__((ext_vector_type(8))) float v8f;

// Types matching the async-to-LDS builtin signature revealed by the compiler:
// parameters are pointers to 16-byte int vectors (global AS1 src, LDS AS3 dst).
typedef int a_v4i __attribute__((vector_size(16)));
typedef __attribute__((address_space(1))) a_v4i* glb_v4i_p;
typedef __attribute__((address_space(3))) a_v4i* lds_v4i_p;

// Row stride 80 floats: 80*4=320B (16B aligned for async B128 writes) and
// 80 % 64 banks == 16, so consecutive rows read by the two half-waves in the
// WMMA B-fragment fetch land on disjoint bank ranges.
#define LDS_STRIDE 80

// Async global->LDS copy (gfx1250 ASYNCcnt path) with safe fallback.
__device__ __forceinline__ void copy16_to_lds(const float* gsrc, float* lds_dst) {
#if __has_builtin(__builtin_amdgcn_global_load_async_to_lds_b128)
    __builtin_amdgcn_global_load_async_to_lds_b128(
        (glb_v4i_p)gsrc, (lds_v4i_p)lds_dst, /*offset=*/0, /*cpol=*/0);
#else
    const float4 v = *(const float4*)gsrc;
    lds_dst[0] = v.x; lds_dst[1] = v.y; lds_dst[2] = v.z; lds_dst[3] = v.w;
#endif
}

__device__ __forceinline__ void wait_async_copies() {
#if __has_builtin(__builtin_amdgcn_s_wait_asynccnt)
    __builtin_amdgcn_s_wait_asynccnt(0);
#endif
}

// ---------------------------------------------------------------------------
// Kernel 1: per-batch window stats via banded-matrix WMMA (f32 16x16x4).
//   S = M x X , Q = M x (X.*X)  with M[w,t] = (0 <= t-w < 20)
//   mean = S/20 ; std = sqrt((Q - S^2/20)/19) ; max/min via VALU loop on LDS.
// Block = 256 threads (8 waves); wave (mt,nt) owns output tile w:[mt*16,+16) e:[nt*16,+16).
// ---------------------------------------------------------------------------
__global__ __launch_bounds__(256) void tc_stats_kernel(const float* __restrict__ x,
                                                       float* __restrict__ out) {
    __shared__ float lx[52 * LDS_STRIDE];

    const int b   = blockIdx.x;
    const int tid = threadIdx.x;
    const float* xb = x + (size_t)b * (T_DIM * E_DIM) + BEGIN * E_DIM; // row t=0 == time 15

    // Stage 49 rows x 64 floats into LDS: 784 x 16B async transfers.
    for (int i = tid; i < 49 * 16; i += 256) {
        const int row = i >> 4, q = i & 15;
        copy16_to_lds(xb + row * E_DIM + q * 4, &lx[row * LDS_STRIDE + q * 4]);
    }
    // Zero pad rows 49..51 (K padding for the WMMA loop).
    for (int i = tid; i < 3 * LDS_STRIDE; i += 256) lx[49 * LDS_STRIDE + i] = 0.0f;
    wait_async_copies();
    __syncthreads();

    const int wave   = tid >> 5;
    const int lane   = tid & 31;
    const int mt     = wave >> 2;        // 0..1  (w tile)
    const int nt     = wave & 3;         // 0..3  (e tile)
    const int laneLo = lane & 15;
    const int hi     = lane >> 4;        // 0: lanes 0-15, 1: lanes 16-31
    const int eCol   = nt * 16 + laneLo; // B/C/D column owned by this lane
    const int wRow   = mt * 16 + laneLo; // A row owned by this lane

    v8f S = {};   // sum accumulator
    v8f Q = {};   // sum-of-squares accumulator

    for (int k4 = 0; k4 < 52; k4 += 4) {
        // A fragment (band matrix, generated in registers):
        //   VGPR0: K = k4 + (hi?2:0), VGPR1: K = k4 + (hi?3:1), M = wRow
        const int ta = k4 + (hi ? 2 : 0);
        const int tb = k4 + (hi ? 3 : 1);
        v2f a, bm, bq;
        a.x = (wRow < W_SZ && ta >= wRow && ta < wRow + C_SZ) ? 1.0f : 0.0f;
        a.y = (wRow < W_SZ && tb >= wRow && tb < wRow + C_SZ) ? 1.0f : 0.0f;

        // B fragment from LDS: row K striped across lanes within a VGPR.
        const float b0 = lx[ta * LDS_STRIDE + eCol];
        const float b1 = lx[tb * LDS_STRIDE + eCol];
        bm.x = b0;      bm.y = b1;
        bq.x = b0 * b0; bq.y = b1 * b1;

        S = __builtin_amdgcn_wmma_f32_16x16x4_f32(false, a, false, bm, (short)0, S, false, false);
        Q = __builtin_amdgcn_wmma_f32_16x16x4_f32(false, a, false, bq, (short)0, Q, false, false);
    }

    // C/D layout: VGPR r -> M = r (lanes 0-15) / M = r+8 (lanes 16-31), N = lane%16.
    for (int r = 0; r < 8; ++r) {
        const int w = mt * 16 + r + (hi ? 8 : 0);
        if (w >= W_SZ) continue;
        const float s    = S[r];
        const float q    = Q[r];
        const float mean = s * (1.0f / (float)C_SZ);
        float var = (q - s * s * (1.0f / (float)C_SZ)) * (1.0f / (float)(C_SZ - 1));
        var = var < 0.0f ? 0.0f : var;
        const float sd = sqrtf(var);

        float mx = -INFINITY, mn = INFINITY;
        #pragma unroll
        for (int c = 0; c < C_SZ; ++c) {
            const float v = lx[(w + c) * LDS_STRIDE + eCol];
            mx = fmaxf(mx, v);
            mn = fminf(mn, v);
        }

        const size_t o = ((size_t)b * W_SZ + w) * OUT_F;
        out[o + eCol]              = mean;  // [0,64)
        out[o + 64 + eCol]         = sd;    // [64,128)
        out[o + 192 + eCol]        = mx;    // [192,256)
        out[o + 256 + eCol]        = mn;    // [256,320)
    }
}

// ---------------------------------------------------------------------------
// Kernel 2: transpose the "last" slab into scratch, column-major.
//   Y[b, j] = x[b, 34 + j/64, j%64] = x[b*4096 + 2176 + j],  j in [0,1920)
//   wsT[j * 8192 + b] = Y[b, j]
// ---------------------------------------------------------------------------
__global__ __launch_bounds__(256) void tc_transpose_kernel(const float* __restrict__ x,
                                                           float* __restrict__ wsT) {
    __shared__ float tile[32][33];
    const int jBase = blockIdx.x * 32;   // 60 blocks cover 1920
    const int bBase = blockIdx.y * 32;   // 256 blocks cover 8192
    const int tx = threadIdx.x, ty = threadIdx.y;

    for (int r = ty; r < 32; r += 8)
        tile[r][tx] = x[(size_t)(bBase + r) * (T_DIM * E_DIM) + (LAST_T0 * E_DIM) + jBase + tx];
    __syncthreads();
    for (int r = ty; r < 32; r += 8)
        wsT[(size_t)(jBase + r) * B_DIM + bBase + tx] = tile[tx][r];
}

// ---------------------------------------------------------------------------
// Kernel 3: per-column bitonic sort (8192 keys) -> ranks.
//   order = argsort(-last) stable; rank = inverse perm. Comparator:
//   "a precedes b" iff a.v > b.v || (a.v == b.v && a.idx < b.idx).
// One block per column j = w*64 + e; 64 KB LDS (value + index).
// ---------------------------------------------------------------------------
__global__ __launch_bounds__(1024) void tc_rank_kernel(const float* __restrict__ wsT,
                                                       float* __restrict__ out) {
    __shared__ float sv[B_DIM];
    __shared__ int   si[B_DIM];
    const int j   = blockIdx.x;
    const int w   = j >> 6;
    const int e   = j & 63;
    const int tid = threadIdx.x;
    const float* col = wsT + (size_t)j * B_DIM;

    for (int i = tid; i < B_DIM; i += 1024) { sv[i] = col[i]; si[i] = i; }
    __syncthreads();

    for (int k = 2; k <= B_DIM; k <<= 1) {
        for (int jj = k >> 1; jj > 0; jj >>= 1) {
            for (int i = tid; i < B_DIM; i += 1024) {
                const int ixj = i ^ jj;
                if (ixj > i) {
                    const bool up = ((i & k) == 0);
                    const float va = sv[i],  vb = sv[ixj];
                    const int   ia = si[i],  ib = si[ixj];
                    const bool aFirst = (va > vb) || (va == vb && ia < ib);
                    if (up ? !aFirst : aFirst) {
                        sv[i] = vb; sv[ixj] = va;
                        si[i] = ib; si[ixj] = ia;
                    }
                }
            }
            __syncthreads();
        }
    }

    const float invB = 1.0f / (float)B_DIM;
    for (int p = tid; p < B_DIM; p += 1024) {
        const int b0 = si[p];
        out[((size_t)b0 * W_SZ + w) * OUT_F + 128 + e] = (float)p * invB;
    }
}

// ---------------------------------------------------------------------------
extern "C" void kernel_launch(void* const* d_in, const int* in_sizes, int n_in,
                              void* d_out, int out_size, void* d_ws, size_t ws_size,
                              hipStream_t stream) {
    const float* x  = (const float*)d_in[0];
    float* out      = (float*)d_out;
    float* wsT      = (float*)d_ws;   // needs 1920*8192*4 = ~63 MB

    tc_stats_kernel<<<B_DIM, 256, 0, stream>>>(x, out);

    dim3 tgrid(NCOL / 32, B_DIM / 32);   // (60, 256)
    tc_transpose_kernel<<<tgrid, dim3(32, 8), 0, stream>>>(x, wsT);

    tc_rank_kernel<<<NCOL, 1024, 0, stream>>>(wsT, out);
}